// GCN_50182397886764
// MI455X (gfx1250) — compile-verified
//
#include <hip/hip_runtime.h>

typedef __attribute__((ext_vector_type(2))) float v2f;
typedef __attribute__((ext_vector_type(8))) float v8f;

// Native fp32 global atomic add (relaxed, device scope -> global_atomic_add_f32)
__device__ __forceinline__ void atomAddF(float* p, float v) {
  __hip_atomic_fetch_add(p, v, __ATOMIC_RELAXED, __HIP_MEMORY_SCOPE_AGENT);
}

__global__ void fill_f32(float* __restrict__ p, float v, long long n) {
  long long i = (long long)blockIdx.x * blockDim.x + threadIdx.x;
  if (i < n) p[i] = v;
}

// deg[dst] += 1 over the real edges (deg pre-initialized to 1.0 for self loop)
__global__ void degree_count(const long long* __restrict__ dst,
                             float* __restrict__ deg, long long E) {
  long long e = (long long)blockIdx.x * blockDim.x + threadIdx.x;
  if (e < E) atomAddF(&deg[(size_t)dst[e]], 1.0f);
}

__global__ void to_rsqrt(float* __restrict__ d, long long n) {
  long long i = (long long)blockIdx.x * blockDim.x + threadIdx.x;
  if (i < n) d[i] = rsqrtf(d[i]);
}

// A[i*64+f] = b[f]  (bias folded into scatter destination init)
__global__ void bias_init(float* __restrict__ A, const float* __restrict__ b,
                          long long n64) {
  long long i = (long long)blockIdx.x * blockDim.x + threadIdx.x;
  if (i < n64) A[i] = b[i & 63];
}

// H[N x 64] = act(X[N x K]) @ W[K x 64], act = ReLU if relu_in.
// LDS-staged, zero-padded tiles; one 16x16 fp32 tile per wave via
// V_WMMA_F32_16X16X4_F32; 4 waves cover the 64 output columns.
// KP = K rounded up to multiple of 4; LDS row stride = KP+2 (bank-conflict pad).
template <int K, int KP>
__global__ void __launch_bounds__(128)
gemm_wmma_f32(const float* __restrict__ X, const float* __restrict__ W,
              float* __restrict__ H, int N, int relu_in) {
  constexpr int STR = KP + 2;                 // dword stride (even -> 8B aligned)
  __shared__ float Xs[16 * STR];              // A tile  [row][KP]
  __shared__ float Ws[64 * STR];              // B tile  [col][KP] (transposed)

  const int tid     = threadIdx.x;
  const int lane    = tid & 31;
  const int wave    = tid >> 5;
  const int rowBase = blockIdx.x * 16;
  const int half    = lane >> 4;              // 0 -> K pair {0,1}, 1 -> {2,3}
  const int l15     = lane & 15;

  // stage W transposed + zero pad (threads walk k, global reads strided by 64; tiny: K*64 floats)
  for (int i = tid; i < 64 * KP; i += 128) {
    const int c = i / KP, k = i - c * KP;
    Ws[c * STR + k] = (k < K) ? W[(size_t)k * 64 + c] : 0.0f;
  }
  // stage X tile + optional ReLU + zero pad (row-clamped loads)
  for (int i = tid; i < 16 * KP; i += 128) {
    const int r = i / KP, k = i - r * KP;
    int row = rowBase + r; if (row >= N) row = N - 1;
    float v = (k < K) ? X[(size_t)row * K + k] : 0.0f;
    if (relu_in) v = fmaxf(v, 0.0f);
    Xs[r * STR + k] = v;
  }
  __syncthreads();

  const int col = wave * 16 + l15;
  v8f acc = {};
#pragma unroll
  for (int k = 0; k < KP; k += 4) {
    const int kk = k + half * 2;
    const v2f a = *(const v2f*)&Xs[l15 * STR + kk];   // ds_load_2addr_b64 pairs
    const v2f b = *(const v2f*)&Ws[col * STR + kk];
    acc = __builtin_amdgcn_wmma_f32_16x16x4_f32(false, a, false, b,
                                                (short)0, acc, false, false);
  }

  // Epilogue: uniform full-tile fast path (no per-store exec masking).
  float* __restrict__ Hp = H + (size_t)(rowBase + half * 8) * 64 + col;
  if (rowBase + 16 <= N) {
#pragma unroll
    for (int r = 0; r < 8; ++r)               // C/D layout: VGPR r -> M=r / M=r+8
      Hp[(size_t)r * 64] = acc[r];
  } else {
#pragma unroll
    for (int r = 0; r < 8; ++r) {
      const int row = rowBase + r + half * 8;
      if (row < N) H[(size_t)row * 64 + col] = acc[r];
    }
  }
}

// A[dst][f] += dinv[src]*dinv[dst] * G[src][f] over E edges plus N self loops.
__global__ void scatter_edges(const long long* __restrict__ srcI,
                              const long long* __restrict__ dstI,
                              const float* __restrict__ dinv,
                              const float* __restrict__ G,
                              float* __restrict__ A,
                              long long E, long long N) {
  long long idx   = (long long)blockIdx.x * blockDim.x + threadIdx.x;
  long long total = (E + N) << 6;
  if (idx >= total) return;
  long long e = idx >> 6;
  int f = (int)(idx & 63);
  long long s, d; float nrm;
  if (e < E) { s = srcI[e]; d = dstI[e]; nrm = dinv[s] * dinv[d]; }
  else       { s = d = e - E; float t = dinv[s]; nrm = t * t; }
  atomAddF(&A[(size_t)d * 64 + f], nrm * G[(size_t)s * 64 + f]);
}

__global__ void pool_count(const long long* __restrict__ batch,
                           float* __restrict__ cnt, long long N) {
  long long i = (long long)blockIdx.x * blockDim.x + threadIdx.x;
  if (i < N) atomAddF(&cnt[(size_t)batch[i]], 1.0f);
}

__global__ void pool_sum(const long long* __restrict__ batch,
                         const float* __restrict__ H,
                         float* __restrict__ pool, long long N) {
  long long idx = (long long)blockIdx.x * blockDim.x + threadIdx.x;
  if (idx < (N << 6)) {
    long long i = idx >> 6;
    int f = (int)(idx & 63);
    atomAddF(&pool[(size_t)batch[i] * 64 + f], H[idx]);
  }
}

// out[g][c] = (pool[g]/max(cnt,1)) . lin_W[:,c] + lin_b[c]
__global__ void head_mlp(const float* __restrict__ pool,
                         const float* __restrict__ cnt,
                         const float* __restrict__ lW,
                         const float* __restrict__ lb,
                         float* __restrict__ out, int G) {
  int idx = blockIdx.x * blockDim.x + threadIdx.x;
  if (idx >= G * 2) return;
  int g = idx >> 1, c = idx & 1;
  float inv = 1.0f / fmaxf(cnt[g], 1.0f);
  float acc = 0.0f;
  for (int f = 0; f < 64; ++f)
    acc += pool[(size_t)g * 64 + f] * lW[f * 2 + c];
  out[idx] = acc * inv + lb[c];
}

static inline int cdiv(long long a, long long b) { return (int)((a + b - 1) / b); }

extern "C" void kernel_launch(void* const* d_in, const int* in_sizes, int n_in,
                              void* d_out, int out_size, void* d_ws, size_t ws_size,
                              hipStream_t stream) {
  const float*     x     = (const float*)d_in[0];
  const long long* ei    = (const long long*)d_in[1];   // int64 [2, E]
  const long long* batch = (const long long*)d_in[2];   // int64 [N]
  const float* W1 = (const float*)d_in[3];
  const float* b1 = (const float*)d_in[4];
  const float* W2 = (const float*)d_in[5];
  const float* b2 = (const float*)d_in[6];
  const float* W3 = (const float*)d_in[7];
  const float* b3 = (const float*)d_in[8];
  const float* lW = (const float*)d_in[9];
  const float* lb = (const float*)d_in[10];
  float* out = (float*)d_out;

  const long long N  = in_sizes[0] / 50;
  const long long E  = in_sizes[1] / 2;
  const long long NG = 1000;
  const long long* srcI = ei;          // row 0
  const long long* dstI = ei + E;      // row 1

  float* Bg   = (float*)d_ws;                 // gemm output  [N x 64]
  float* Ba   = Bg + (size_t)N * 64;          // aggregate    [N x 64]
  float* dinv = Ba + (size_t)N * 64;          // deg -> rsqrt [N]
  float* pool = dinv + N;                     // [NG x 64]
  float* cnt  = pool + NG * 64;               // [NG]

  const long long n64 = N * 64;
  const long long sc  = (E + N) * 64;
  const int gemmBlocks = cdiv(N, 16);

  // symmetric normalization: deg over dst incl. self loop, dinv = rsqrt(deg)
  fill_f32<<<cdiv(N, 256), 256, 0, stream>>>(dinv, 1.0f, N);
  degree_count<<<cdiv(E, 256), 256, 0, stream>>>(dstI, dinv, E);
  to_rsqrt<<<cdiv(N, 256), 256, 0, stream>>>(dinv, N);

  // layer 1: h1 = relu(scatter(x @ W1) + b1)   (relu deferred to next gemm)
  gemm_wmma_f32<50, 52><<<gemmBlocks, 128, 0, stream>>>(x, W1, Bg, (int)N, 0);
  bias_init<<<cdiv(n64, 256), 256, 0, stream>>>(Ba, b1, n64);
  scatter_edges<<<cdiv(sc, 256), 256, 0, stream>>>(srcI, dstI, dinv, Bg, Ba, E, N);

  // layer 2
  gemm_wmma_f32<64, 64><<<gemmBlocks, 128, 0, stream>>>(Ba, W2, Bg, (int)N, 1);
  bias_init<<<cdiv(n64, 256), 256, 0, stream>>>(Ba, b2, n64);
  scatter_edges<<<cdiv(sc, 256), 256, 0, stream>>>(srcI, dstI, dinv, Bg, Ba, E, N);

  // layer 3 (no relu after)
  gemm_wmma_f32<64, 64><<<gemmBlocks, 128, 0, stream>>>(Ba, W3, Bg, (int)N, 1);
  bias_init<<<cdiv(n64, 256), 256, 0, stream>>>(Ba, b3, n64);
  scatter_edges<<<cdiv(sc, 256), 256, 0, stream>>>(srcI, dstI, dinv, Bg, Ba, E, N);

  // global mean pool + linear head
  fill_f32<<<cdiv(NG * 64, 256), 256, 0, stream>>>(pool, 0.0f, NG * 64);
  fill_f32<<<cdiv(NG, 256), 256, 0, stream>>>(cnt, 0.0f, NG);
  pool_count<<<cdiv(N, 256), 256, 0, stream>>>(batch, cnt, N);
  pool_sum<<<cdiv(n64, 256), 256, 0, stream>>>(batch, Ba, pool, N);
  head_mlp<<<cdiv(NG * 2, 256), 256, 0, stream>>>(pool, cnt, lW, lb, out, (int)NG);
}